// MultiTimeAttention_56143812493710
// MI455X (gfx1250) — compile-verified
//
#include <hip/hip_runtime.h>

typedef __attribute__((ext_vector_type(2))) float v2f;
typedef __attribute__((ext_vector_type(8))) float v8f;

#define BB  8
#define LQ  128
#define LK  256
#define DIM 64
#define ET  128
#define NH  8
#define ETK 16

// ---------------------------------------------------------------------------
// Generic projection: Y(Mx128) = X(MxK) @ W(Kx128) + bias, f32 WMMA 16x16x4.
// One 16x16 output tile per wave; 4 waves per block.
// A-frag (16x4 f32): lanes 0-15 hold K={0,1}, lanes 16-31 hold K={2,3}.
// B-frag (4x16 f32): mirrored; C/D: vgpr r -> row r (lanes 0-15) / r+8 (16-31).
// ---------------------------------------------------------------------------
__global__ __launch_bounds__(128, 1)
void mta_proj_f32wmma(const float* __restrict__ X, const float* __restrict__ W,
                      const float* __restrict__ bias, float* __restrict__ Y,
                      int M, int K) {
  const int lane = threadIdx.x & 31;
  const int wv   = threadIdx.x >> 5;
  const int tile = blockIdx.x * 4 + wv;
  const int trow = tile >> 3;            // 128/16 = 8 column tiles
  const int tcol = tile & 7;
  const int l16  = lane & 15;
  const int klo  = (lane >> 4) * 2;

  const float* arow = X + (size_t)(trow * 16 + l16) * K;
  const float* wcol = W + tcol * 16 + l16;

  v8f acc = {};
  const int ksteps = K >> 2;
  for (int t = 0; t < ksteps; ++t) {
    const int k = 4 * t + klo;
    v2f a = *(const v2f*)(arow + k);
    v2f b;
    b.x = wcol[(size_t)k * 128];
    b.y = wcol[(size_t)(k + 1) * 128];
    acc = __builtin_amdgcn_wmma_f32_16x16x4_f32(false, a, false, b, (short)0,
                                                acc, false, false);
  }

  const int   col = tcol * 16 + l16;
  const float bs  = bias[col];
  const int   m0  = (lane < 16) ? 0 : 8;
#pragma unroll
  for (int r = 0; r < 8; ++r)
    Y[(size_t)(trow * 16 + m0 + r) * 128 + col] = acc[r] + bs;
}

// ---------------------------------------------------------------------------
// Fused per-(b,h) attention:
//   S = Q_h K_h^T / 4  (128x256), E = exp(S - rowmax)
//   numer = E @ (mask*value), denom = E @ mask   (both 128x64x256 GEMMs)
//   x[q,d] = numer/denom -> xb[b][q][h*64+d]
// Block = 256 threads (8 waves); wave w owns query rows 16w..16w+15.
// E strip (16x256) lives in VGPRs; K_h staged in LDS; E round-trips through a
// 4KB/wave LDS scratch in 64-wide K chunks for D->A fragment re-layout.
// ---------------------------------------------------------------------------
__global__ __launch_bounds__(256, 1)
void mta_attention(const float* __restrict__ qp, const float* __restrict__ kp,
                   const float* __restrict__ value, const int* __restrict__ mask,
                   float* __restrict__ xb) {
  __shared__ float kls[LK * ETK];       // K_h 256x16 f32 = 16 KB
  __shared__ float els[8 * 16 * 64];    // 8 waves x (16x64) E chunk = 32 KB

  const int b    = blockIdx.x >> 3;
  const int h    = blockIdx.x & 7;
  const int lane = threadIdx.x & 31;
  const int wv   = threadIdx.x >> 5;
  const int l16  = lane & 15;
  const int klo  = (lane >> 4) * 2;

  // Cooperative load of K_h (256 rows x 16 cols) into LDS: one row per thread.
  {
    const int krow = threadIdx.x;       // 0..255
    const float4* src = (const float4*)(kp + (size_t)(b * LK + krow) * ET + h * ETK);
    float4* dst = (float4*)(kls + krow * ETK);
#pragma unroll
    for (int i = 0; i < 4; ++i) dst[i] = src[i];
  }
  __syncthreads();

  // Q fragments for this wave's 16 rows (16x16, 4 k-steps of 4).
  v2f qf[4];
  {
    const float* qrow = qp + (size_t)(b * LQ + wv * 16 + l16) * ET + h * ETK;
#pragma unroll
    for (int t = 0; t < 4; ++t)
      qf[t] = *(const v2f*)(qrow + 4 * t + klo);
  }

  // Phase 1: S strip = Q (16x16) @ K^T (16x256) as 16 WMMA tiles in registers.
  v8f sacc[16];
#pragma unroll
  for (int j = 0; j < 16; ++j) {
    v8f c = {};
#pragma unroll
    for (int t = 0; t < 4; ++t) {
      v2f bf = *(const v2f*)(kls + (j * 16 + l16) * ETK + 4 * t + klo);
      c = __builtin_amdgcn_wmma_f32_16x16x4_f32(false, qf[t], false, bf, (short)0,
                                                c, false, false);
    }
    sacc[j] = c;
  }

  // Scale (1/sqrt(16)), row max across the 256 keys, then exp.
  float mr[8];
#pragma unroll
  for (int r = 0; r < 8; ++r) mr[r] = -3.0e38f;
#pragma unroll
  for (int j = 0; j < 16; ++j)
#pragma unroll
    for (int r = 0; r < 8; ++r) {
      float s = sacc[j][r] * 0.25f;
      sacc[j][r] = s;
      mr[r] = fmaxf(mr[r], s);
    }
  // Reduce across the 16 lanes of each half-wave (xor of bits 0..3 stays in half).
#pragma unroll
  for (int m = 1; m <= 8; m <<= 1)
#pragma unroll
    for (int r = 0; r < 8; ++r)
      mr[r] = fmaxf(mr[r], __shfl_xor(mr[r], m, 32));
#pragma unroll
  for (int j = 0; j < 16; ++j)
#pragma unroll
    for (int r = 0; r < 8; ++r)
      sacc[j][r] = __expf(sacc[j][r] - mr[r]);

  // Phase 2: numer = E @ (mask*value), denom = E @ mask, K chunked by 64.
  v8f num[4], den[4];
  {
    v8f z = {};
#pragma unroll
    for (int j = 0; j < 4; ++j) { num[j] = z; den[j] = z; }
  }
  float* myE = els + wv * (16 * 64);
  const int rbase = (lane < 16) ? 0 : 8;
  const int*   mbase = mask  + (size_t)b * LK * DIM;
  const float* vbase = value + (size_t)b * LK * DIM;

#pragma unroll
  for (int kb = 0; kb < 4; ++kb) {
    // Spill this K-chunk of E (D-layout) to wave-private LDS.
#pragma unroll
    for (int jj = 0; jj < 4; ++jj) {
      const int j = kb * 4 + jj;
#pragma unroll
      for (int r = 0; r < 8; ++r)
        myE[(rbase + r) * 64 + jj * 16 + l16] = sacc[j][r];
    }
    asm volatile("s_wait_dscnt 0" ::: "memory");  // LDS RAW within wave

#pragma unroll 4
    for (int t = 0; t < 16; ++t) {
      // A-fragment of E: row = l16, cols = 4t+klo..+1 (chunk-local K).
      v2f af = *(const v2f*)(myE + l16 * 64 + 4 * t + klo);
      const int kg = kb * 64 + 4 * t + klo;       // global K index
      const int*   m0p = mbase + (size_t)kg * DIM;
      const float* v0p = vbase + (size_t)kg * DIM;
#pragma unroll
      for (int j = 0; j < 4; ++j) {
        const int d   = j * 16 + l16;
        const int   mk0 = m0p[d];
        const int   mk1 = m0p[d + DIM];
        const float vv0 = v0p[d];
        const float vv1 = v0p[d + DIM];
        v2f bpv, bpm;
        bpv.x = mk0 ? vv0 : 0.0f;
        bpv.y = mk1 ? vv1 : 0.0f;
        bpm.x = mk0 ? 1.0f : 0.0f;
        bpm.y = mk1 ? 1.0f : 0.0f;
        num[j] = __builtin_amdgcn_wmma_f32_16x16x4_f32(false, af, false, bpv,
                                                       (short)0, num[j], false, false);
        den[j] = __builtin_amdgcn_wmma_f32_16x16x4_f32(false, af, false, bpm,
                                                       (short)0, den[j], false, false);
      }
    }
  }

  // x = numer/denom -> merged-head layout xb[b][q][h*64 + d].
#pragma unroll
  for (int j = 0; j < 4; ++j)
#pragma unroll
    for (int r = 0; r < 8; ++r) {
      const int q = wv * 16 + rbase + r;
      const int c = h * DIM + j * 16 + l16;
      xb[((size_t)(b * LQ + q)) * (NH * DIM) + c] =
          num[j][r] / fmaxf(den[j][r], 1e-30f);
    }
}

// ---------------------------------------------------------------------------
extern "C" void kernel_launch(void* const* d_in, const int* in_sizes, int n_in,
                              void* d_out, int out_size, void* d_ws, size_t ws_size,
                              hipStream_t stream) {
  (void)in_sizes; (void)n_in; (void)out_size; (void)ws_size;

  const float* query = (const float*)d_in[0];   // (8,128,128)
  const float* key   = (const float*)d_in[1];   // (8,256,128)
  const float* value = (const float*)d_in[2];   // (8,256,64)
  const int*   maskp = (const int*)  d_in[3];   // (8,256,64)
  const float* Wq    = (const float*)d_in[4];   // (128,128)
  const float* bq    = (const float*)d_in[5];
  const float* Wk    = (const float*)d_in[6];
  const float* bk    = (const float*)d_in[7];
  const float* Wo    = (const float*)d_in[8];   // (512,128)
  const float* bo    = (const float*)d_in[9];
  float* out = (float*)d_out;                   // (8,128,128)

  // Workspace: q_proj (1024x128) | k_proj (2048x128) | merged x (1024x512)
  float* qp = (float*)d_ws;
  float* kp = qp + 1024 * 128;
  float* xb = kp + 2048 * 128;

  // q/k projections (independent, same stream keeps pipeline order for attn).
  mta_proj_f32wmma<<<128, 128, 0, stream>>>(query, Wq, bq, qp, 1024, 128);
  mta_proj_f32wmma<<<256, 128, 0, stream>>>(key,   Wk, bk, kp, 2048, 128);

  // Fused masked attention per (b,h): 64 blocks x 8 waves.
  mta_attention<<<64, 256, 0, stream>>>(qp, kp, value, maskp, xb);

  // Output projection: (1024x512) @ (512x128) + bo -> d_out.
  mta_proj_f32wmma<<<128, 128, 0, stream>>>(xb, Wo, bo, out, 1024, 512);
}